// ProbAttention_38723425141433
// MI455X (gfx1250) — compile-verified
//
#include <hip/hip_runtime.h>
#include <hip/hip_bf16.h>

// ProbSparse attention (Informer) for MI455X / gfx1250, wave32.
// B=16, L=1024, D=64, u = sample_k = 64.
// f32 everywhere (memory-bound; ~17MB fits in L2), V_WMMA_F32_16X16X4_F32 for
// the dense GEMMs, Tensor Data Mover (TENSOR_LOAD_TO_LDS) for tile staging.

#define B_   16
#define L_   1024
#define D_   64
#define U_   64
#define SK_  64
#define NT_  (L_ / 16)     // 64 key tiles in the dense phase
#define CH_  16            // 16 K-chunks of 64 rows in the sampling phase

typedef float        v2f __attribute__((ext_vector_type(2)));
typedef float        v8f __attribute__((ext_vector_type(8)));
typedef unsigned int v4u __attribute__((ext_vector_type(4)));
typedef int          v8i __attribute__((ext_vector_type(8)));
typedef int          v4i __attribute__((ext_vector_type(4)));

// ---------------------------------------------------------------------------
// TDM: load a (rows x 64) f32 tile from global into LDS with 1-dword padding
// every 64 dwords (LDS row pitch = 65 dwords -> conflict-free ds_loads).
// Descriptor layout per CDNA5 ISA. Uniform operands only.
// ---------------------------------------------------------------------------
__device__ __forceinline__ void tdm_load_rows_x64(const float* gsrc,
                                                  unsigned lds_byte_off,
                                                  unsigned rows) {
    unsigned long long ga = (unsigned long long)(const void*)gsrc;
    v4u g0;
    g0[0] = 1u;                                          // count=1 (valid user D#)
    g0[1] = lds_byte_off;                                // lds_addr
    g0[2] = (unsigned)(ga & 0xffffffffu);                // global_addr[31:0]
    g0[3] = (unsigned)((ga >> 32) & 0x1ffffffu)          // global_addr[56:32]
          | (2u << 30);                                  // type=2 ("image")
    // group1 (256b): data_size=2 (4B) | pad_enable | pad_interval=5 (64 dw),
    // pad_amount=0 (1 dw); tensor_dim0=64, tensor_dim1=1024, tile_dim0=64,
    // tile_dim1=rows, tensor_dim0_stride=64.
    unsigned long long q0 = 0x01520000ull | (64ull << 48);
    unsigned long long q1 = (1024ull << 16) | (64ull << 48);
    unsigned long long q2 = (unsigned long long)rows | (64ull << 32);
    unsigned long long q3 = 0ull;
    v8i g1;
    g1[0] = (int)(unsigned)q0;  g1[1] = (int)(unsigned)(q0 >> 32);
    g1[2] = (int)(unsigned)q1;  g1[3] = (int)(unsigned)(q1 >> 32);
    g1[4] = (int)(unsigned)q2;  g1[5] = (int)(unsigned)(q2 >> 32);
    g1[6] = (int)(unsigned)q3;  g1[7] = (int)(unsigned)(q3 >> 32);
    v4i z = {};
#if __clang_major__ >= 23
    v8i z8 = {};
    __builtin_amdgcn_tensor_load_to_lds(g0, g1, z, z, z8, 0);
#else
    __builtin_amdgcn_tensor_load_to_lds(g0, g1, z, z, 0);
#endif
}

#define LDS_OFF(p) ((unsigned)(unsigned long long)(const void*)(p))

// ---------------------------------------------------------------------------
// Kernel A: M[b,l] = max_s(q.k_idx) - sum_s(q.k_idx)/L.
// Block = 256 threads = 256 queries of one batch; K[b] streamed through LDS in
// 16 double-buffered 64-row chunks via TDM; sample gathers hit LDS, not L2.
// ---------------------------------------------------------------------------
__global__ __launch_bounds__(256) void pa_compute_M(
    const float* __restrict__ Q, const float* __restrict__ K,
    const int* __restrict__ idx, float* __restrict__ Mscore)
{
    __shared__ float kch[2][64][65];                     // 2 x 16.25KB

    const int b   = blockIdx.x >> 2;
    const int lc  = blockIdx.x & 3;
    const int l   = lc * 256 + threadIdx.x;
    const int gid = b * L_ + l;
    const bool w0 = (threadIdx.x >> 5) == 0;

    float qr[D_];
    const float* q = Q + (size_t)gid * D_;
#pragma unroll
    for (int d = 0; d < D_; ++d) qr[d] = q[d];

    const float* kb0 = K + (size_t)b * L_ * D_;
    if (w0) tdm_load_rows_x64(kb0, LDS_OFF(&kch[0][0][0]), 64);

    float mx = -1e30f, sm = 0.0f;
    for (int ch = 0; ch < CH_; ++ch) {
        __syncthreads();                                 // prev reads of buf[(ch+1)&1] done
        if (w0) {
            if (ch + 1 < CH_) {
                tdm_load_rows_x64(kb0 + (size_t)(ch + 1) * 64 * D_,
                                  LDS_OFF(&kch[(ch + 1) & 1][0][0]), 64);
                __builtin_amdgcn_s_wait_tensorcnt(1);    // chunk ch resident
            } else {
                __builtin_amdgcn_s_wait_tensorcnt(0);
            }
        }
        __syncthreads();                                 // chunk ch visible to all waves
        const float* kc = &kch[ch & 1][0][0];
        for (int s = 0; s < SK_; ++s) {
            int j = idx[l * SK_ + s];
            if ((j >> 6) == ch) {
                const float* kr = kc + (size_t)(j & 63) * 65;
                float acc = 0.0f;
#pragma unroll
                for (int d = 0; d < D_; ++d) acc = fmaf(qr[d], kr[d], acc);
                mx = fmaxf(mx, acc);
                sm += acc;
            }
        }
    }
    Mscore[gid] = mx - sm * (1.0f / (float)L_);
}

// ---------------------------------------------------------------------------
// Kernel B: top-64 indices per batch (descending, like lax.top_k)
// ---------------------------------------------------------------------------
__global__ __launch_bounds__(256) void pa_topk64(
    const float* __restrict__ Mscore, int* __restrict__ Mtop)
{
    __shared__ float sv[L_];
    __shared__ float rmax[256];
    __shared__ int   ridx[256];
    const int b = blockIdx.x, t = threadIdx.x;

    for (int i = t; i < L_; i += 256) sv[i] = Mscore[b * L_ + i];
    __syncthreads();

    for (int pick = 0; pick < U_; ++pick) {
        float best = -1e31f; int bi = 0;
        for (int i = t; i < L_; i += 256) {
            float v = sv[i];
            if (v > best) { best = v; bi = i; }
        }
        rmax[t] = best; ridx[t] = bi;
        __syncthreads();
        for (int s = 128; s > 0; s >>= 1) {
            if (t < s && rmax[t + s] > rmax[t]) {
                rmax[t] = rmax[t + s]; ridx[t] = ridx[t + s];
            }
            __syncthreads();
        }
        if (t == 0) { Mtop[b * U_ + pick] = ridx[0]; sv[ridx[0]] = -1e31f; }
        __syncthreads();
    }
}

// ---------------------------------------------------------------------------
// Kernel C: context = cumsum(values, axis=1); one thread per (b,d) column
// ---------------------------------------------------------------------------
__global__ __launch_bounds__(256) void pa_cumsum(
    const float* __restrict__ V, float* __restrict__ Out)
{
    int gid = blockIdx.x * blockDim.x + threadIdx.x;     // b*D + d
    if (gid >= B_ * D_) return;
    int b = gid / D_, d = gid % D_;
    const float* vp = V + (size_t)b * L_ * D_ + d;
    float* op = Out + (size_t)b * L_ * D_ + d;
    float acc = 0.0f;
    for (int l = 0; l < L_; ++l) {
        acc += vp[(size_t)l * D_];
        op[(size_t)l * D_] = acc;
    }
}

// ---------------------------------------------------------------------------
// Kernel D: dense attention for the 64 selected queries per batch.
// Grid = B, block = 128 (4 waves, one 16-row M-tile each). K/V tiles staged in
// LDS by TDM (double-buffered); flash-attention online softmax; both GEMMs via
// V_WMMA_F32_16X16X4_F32.
//
// Fragment layouts (ISA 7.12.2, wave32):
//   A 16x4 : lane = 16*(k>=2) + m, vgpr = k&1
//   B 4x16 : lane = 16*(k>=2) + n, vgpr = k&1
//   C 16x16: vgpr r, lanes 0-15 -> (M=r, N=lane), lanes 16-31 -> (M=r+8, N=lane-16)
// ---------------------------------------------------------------------------
__global__ __launch_bounds__(128) void pa_core(
    const float* __restrict__ Q, const float* __restrict__ K,
    const float* __restrict__ V, const int* __restrict__ Mtop,
    float* __restrict__ Out)
{
    __shared__ float kbuf[2][16][65];                    // TDM-padded tiles
    __shared__ float vbuf[2][16][65];
    __shared__ float p_lds[4][16][17];                   // per-wave P staging

    const int b    = blockIdx.x;
    const int wave = threadIdx.x >> 5;
    const int lane = threadIdx.x & 31;
    const int hi   = lane >> 4;
    const int ln   = lane & 15;
    const int kb2  = hi * 2;
    const bool w0  = (wave == 0);

    int trow[8];
#pragma unroll
    for (int r = 0; r < 8; ++r)
        trow[r] = Mtop[b * U_ + wave * 16 + r + 8 * hi];

    v2f a_q[16];
    {
        int t = Mtop[b * U_ + wave * 16 + ln];           // A-row m == ln
        const float* qrow = Q + ((size_t)b * L_ + t) * D_;
#pragma unroll
        for (int ks = 0; ks < 16; ++ks) {
            a_q[ks][0] = qrow[ks * 4 + kb2 + 0];
            a_q[ks][1] = qrow[ks * 4 + kb2 + 1];
        }
    }

    float m_i[8], l_i[8];
    v8f o_acc[4];
    const v8f vzero = {};
#pragma unroll
    for (int r = 0; r < 8; ++r) { m_i[r] = -1e30f; l_i[r] = 0.0f; }
#pragma unroll
    for (int dt = 0; dt < 4; ++dt) o_acc[dt] = vzero;

    const float scale = 0.125f;
    const float* kbase = K + (size_t)b * L_ * D_;
    const float* vbase = V + (size_t)b * L_ * D_;

    if (w0) {                                            // prologue: tile 0
        tdm_load_rows_x64(kbase, LDS_OFF(&kbuf[0][0][0]), 16);
        tdm_load_rows_x64(vbase, LDS_OFF(&vbuf[0][0][0]), 16);
    }

    for (int nt = 0; nt < NT_; ++nt) {
        __syncthreads();                                 // reads of buf[(nt+1)&1] done
        if (w0) {
            if (nt + 1 < NT_) {
                const int nb = (nt + 1) & 1;
                tdm_load_rows_x64(kbase + (size_t)(nt + 1) * 16 * D_,
                                  LDS_OFF(&kbuf[nb][0][0]), 16);
                tdm_load_rows_x64(vbase + (size_t)(nt + 1) * 16 * D_,
                                  LDS_OFF(&vbuf[nb][0][0]), 16);
                __builtin_amdgcn_s_wait_tensorcnt(2);    // tile nt resident
            } else {
                __builtin_amdgcn_s_wait_tensorcnt(0);
            }
        }
        __syncthreads();                                 // tile nt visible

        const float* kf = &kbuf[nt & 1][0][0];
        const float* vf = &vbuf[nt & 1][0][0];

        // ---- S = Qr(16x64) * K_tile^T(64x16): 16 x wmma f32 16x16x4 ----
        v8f c = vzero;
#pragma unroll
        for (int ks = 0; ks < 16; ++ks) {
            v2f bf;                                      // B[k][n] = K[n][4ks+k]
            bf[0] = kf[(size_t)ln * 65 + ks * 4 + kb2 + 0];
            bf[1] = kf[(size_t)ln * 65 + ks * 4 + kb2 + 1];
            c = __builtin_amdgcn_wmma_f32_16x16x4_f32(
                    false, a_q[ks], false, bf, (short)0, c, false, false);
        }

        // ---- scale + block-causal mask: masked iff (col>>3) > (row>>3) ----
        const int col = nt * 16 + ln;
#pragma unroll
        for (int r = 0; r < 8; ++r) {
            float s = c[r] * scale;
            if ((col >> 3) > (trow[r] >> 3)) s = -1e30f;
            c[r] = s;
        }

        // ---- row max across each 16-lane half ----
        float pmax[8];
#pragma unroll
        for (int r = 0; r < 8; ++r) pmax[r] = c[r];
#pragma unroll
        for (int off = 1; off < 16; off <<= 1)
#pragma unroll
            for (int r = 0; r < 8; ++r)
                pmax[r] = fmaxf(pmax[r], __shfl_xor(pmax[r], off, 32));

        // ---- online softmax update ----
        float corr[8], psum[8];
#pragma unroll
        for (int r = 0; r < 8; ++r) {
            float mnew = fmaxf(m_i[r], pmax[r]);
            corr[r] = __expf(m_i[r] - mnew);
            m_i[r]  = mnew;
            float p = __expf(c[r] - mnew);
            c[r]    = p;
            psum[r] = p;
        }
#pragma unroll
        for (int off = 1; off < 16; off <<= 1)
#pragma unroll
            for (int r = 0; r < 8; ++r)
                psum[r] += __shfl_xor(psum[r], off, 32);
#pragma unroll
        for (int r = 0; r < 8; ++r) l_i[r] = l_i[r] * corr[r] + psum[r];
#pragma unroll
        for (int dt = 0; dt < 4; ++dt)
#pragma unroll
            for (int r = 0; r < 8; ++r) o_acc[dt][r] *= corr[r];

        // ---- C-layout -> A-layout via per-wave LDS pad ----
#pragma unroll
        for (int r = 0; r < 8; ++r)
            p_lds[wave][r + 8 * hi][ln] = c[r];
        __syncthreads();

        // ---- O += P(16x16) * V_tile(16x64): 4 k-steps x 4 D-tiles ----
#pragma unroll
        for (int ks = 0; ks < 4; ++ks) {
            v2f ap;                                      // A-row m == ln
            ap[0] = p_lds[wave][ln][ks * 4 + kb2 + 0];
            ap[1] = p_lds[wave][ln][ks * 4 + kb2 + 1];
#pragma unroll
            for (int dt = 0; dt < 4; ++dt) {
                v2f bv;                                  // B[k][n] = V[4ks+k][dt*16+n]
                bv[0] = vf[(size_t)(ks * 4 + kb2 + 0) * 65 + dt * 16 + ln];
                bv[1] = vf[(size_t)(ks * 4 + kb2 + 1) * 65 + dt * 16 + ln];
                o_acc[dt] = __builtin_amdgcn_wmma_f32_16x16x4_f32(
                                false, ap, false, bv, (short)0, o_acc[dt],
                                false, false);
            }
        }
    }

    // ---- normalize and scatter into the cumsum context ----
#pragma unroll
    for (int dt = 0; dt < 4; ++dt) {
        const int d = dt * 16 + ln;
#pragma unroll
        for (int r = 0; r < 8; ++r) {
            float o = o_acc[dt][r] / l_i[r];
            Out[((size_t)b * L_ + trow[r]) * D_ + d] = o;
        }
    }
}

// ---------------------------------------------------------------------------
extern "C" void kernel_launch(void* const* d_in, const int* in_sizes, int n_in,
                              void* d_out, int out_size, void* d_ws, size_t ws_size,
                              hipStream_t stream) {
    const float* Q   = (const float*)d_in[0];
    const float* K   = (const float*)d_in[1];
    const float* V   = (const float*)d_in[2];
    const int*   idx = (const int*)d_in[3];
    // d_in[4] = mask: block-causal, recomputed arithmetically in pa_core.
    float* Out = (float*)d_out;

    float* Mscore = (float*)d_ws;                                         // B*L f32
    int*   Mtop   = (int*)((char*)d_ws + (size_t)B_ * L_ * sizeof(float)); // B*U i32

    pa_compute_M<<<B_ * 4, 256, 0, stream>>>(Q, K, idx, Mscore);
    pa_topk64  <<<B_, 256, 0, stream>>>(Mscore, Mtop);
    pa_cumsum  <<<(B_ * D_ + 255) / 256, 256, 0, stream>>>(V, Out);
    pa_core    <<<B_, 128, 0, stream>>>(Q, K, V, Mtop, Out);
}